// HyperbolicPersistentHomology_16870631539343
// MI455X (gfx1250) — compile-verified
//
#include <hip/hip_runtime.h>
#include <stdint.h>

// Problem constants (from setup_inputs): emb (8, 16384, 256) f32, L = 64.
#define B_ 8
#define N_ 16384
#define D_ 256
#define L_ 64

typedef float v2f __attribute__((ext_vector_type(2)));
typedef float v8f __attribute__((ext_vector_type(8)));
typedef uint32_t u32x4 __attribute__((ext_vector_type(4)));
typedef uint32_t u32x8 __attribute__((ext_vector_type(8)));

// ---------------------------------------------------------------------------
// Kernel 1: per-row squared norms xn[b*N + r] = sum_d emb[b,r,d]^2
// ---------------------------------------------------------------------------
__global__ __launch_bounds__(256) void k_rownorm(const float* __restrict__ emb,
                                                 float* __restrict__ xn) {
    int row = blockIdx.x * 256 + threadIdx.x;          // 0 .. B*N-1
    const float4* p = (const float4*)(emb + (size_t)row * D_);
    float s = 0.f;
#pragma unroll 8
    for (int i = 0; i < D_ / 4; ++i) {
        float4 v = p[i];
        s += v.x * v.x + v.y * v.y + v.z * v.z + v.w * v.w;
    }
    xn[row] = s;
}

// ---------------------------------------------------------------------------
// Kernel 2: init edge-weight matrix to +inf (ordered-uint encoding)
// ---------------------------------------------------------------------------
__global__ void k_init_ew(uint32_t* __restrict__ ew) {
    int i = blockIdx.x * 256 + threadIdx.x;
    if (i < B_ * L_ * L_) ew[i] = 0x7F800000u;         // +inf bits
}

// ---------------------------------------------------------------------------
// Kernel 3: MaxMin landmark selection (one block per batch, sequential in k).
// min-dist kept in registers (16 rows / thread). Gathers landmark vectors,
// norms and indices into workspace for the witness GEMM.
// ---------------------------------------------------------------------------
__global__ __launch_bounds__(1024) void k_select(const float* __restrict__ emb,
                                                 const float* __restrict__ xn,
                                                 const float* __restrict__ csp,
                                                 float* __restrict__ lmvecs,
                                                 float* __restrict__ lnw,
                                                 int* __restrict__ idxArr) {
    const int b = blockIdx.x;
    const int t = threadIdx.x;
    const float c = fabsf(csp[0]);
    __shared__ float lmS[D_];
    __shared__ float redV[1024];
    __shared__ int   redI[1024];
    __shared__ int   prevS;

    const float* embB = emb + (size_t)b * N_ * D_;
    const float* xnB  = xn + (size_t)b * N_;

    float md[16];
#pragma unroll
    for (int i = 0; i < 16; ++i) md[i] = __builtin_inff();

    int prev = 0;                                      // reference seeds with 0
    for (int k = 0; k < L_; ++k) {
        if (t < D_) {
            float v = embB[(size_t)prev * D_ + t];
            lmS[t] = v;
            lmvecs[((size_t)b * L_ + k) * D_ + t] = v;
        }
        if (t == 0) {
            lnw[b * L_ + k]    = xnB[prev];
            idxArr[b * L_ + k] = prev;
        }
        __syncthreads();
        if (k == L_ - 1) break;

        const float lnv = xnB[prev];
        float bestV = -1.0f;
        int   bestI = 0;
#pragma unroll
        for (int i = 0; i < 16; ++i) {
            const int r = t + i * 1024;
            const float4* p = (const float4*)(embB + (size_t)r * D_);
            const float4* q = (const float4*)lmS;
            float dot = 0.f;
#pragma unroll 8
            for (int j = 0; j < D_ / 4; ++j) {
                float4 a = p[j], bb = q[j];
                dot += a.x * bb.x + a.y * bb.y + a.z * bb.z + a.w * bb.w;
            }
            const float xr   = xnB[r];
            float diff = fmaxf(xr + lnv - 2.f * dot, 1e-10f);
            float fac  = fmaxf(fmaf(c, xr + lnv, 1.0f), 1e-6f);
            float dist = sqrtf(diff) * sqrtf(fac);
            float m = fminf(md[i], dist);
            md[i] = m;
            if (m > bestV || (m == bestV && r < bestI)) { bestV = m; bestI = r; }
        }
        redV[t] = bestV; redI[t] = bestI;
        __syncthreads();
        // deterministic tree argmax, jnp.argmax tiebreak: smallest index wins
        for (int s = 512; s > 0; s >>= 1) {
            if (t < s) {
                float v2 = redV[t + s]; int i2 = redI[t + s];
                if (v2 > redV[t] || (v2 == redV[t] && i2 < redI[t])) {
                    redV[t] = v2; redI[t] = i2;
                }
            }
            __syncthreads();
        }
        if (t == 0) prevS = redI[0];
        __syncthreads();
        prev = prevS;
    }
}

// ---------------------------------------------------------------------------
// Kernel 4: witness distances via f32 WMMA GEMM (X · Lmᵀ, K=256), fused
// top-2 landmark selection per point + ordered-uint atomicMin scatter.
// Block: 256 threads = 8 waves, 16 rows/wave -> 128 rows/block.
// Landmark tile (64x256 f32, 64 KB) is staged into LDS by the Tensor Data
// Mover (tensor_load_to_lds, 2-D descriptor, 2 SGPR groups).
// ---------------------------------------------------------------------------
__global__ __launch_bounds__(256) void k_witness(const float* __restrict__ emb,
                                                 const float* __restrict__ xn,
                                                 const float* __restrict__ lmvecs,
                                                 const float* __restrict__ lnw,
                                                 const float* __restrict__ csp,
                                                 uint32_t* __restrict__ ew) {
    __shared__ float lmS[L_ * D_];                     // 64 KB landmark tile
    const int b = blockIdx.y;
    const int t = threadIdx.x;
    const float c = fabsf(csp[0]);

    // ---- TDM stage: global lmvecs[b] (64x256 f32, contiguous) -> LDS ------
    if (t < 32) {                                      // wave 0 issues the DMA
        const uint64_t gaddr = (uint64_t)(uintptr_t)(lmvecs + (size_t)b * L_ * D_);
        const uint32_t ldsa  = (uint32_t)(uintptr_t)&lmS[0];   // flat LDS: addr[31:0]

        // D# group 0 (128b): count=1 | lds_addr | global_addr[56:0] | type=2
        u32x4 g0;
        g0.x = 1u;                                     // count=1, user mode
        g0.y = ldsa;                                   // lds_addr (bytes)
        g0.z = (uint32_t)gaddr;                        // global_addr[31:0]
        g0.w = (uint32_t)(gaddr >> 32) | (2u << 30);   // global_addr[56:32], type=2

        // D# group 1 (256b): wg_mask=0, data_size=4B, no barrier/iterate/pad,
        // tensor_dim0=256, tensor_dim1=64, tile_dim0=256, tile_dim1=64,
        // tensor_dim0_stride=256 (contiguous rows).
        u32x8 g1;
        g1.s0 = (2u << 16);                            // data_size = 2 (4 bytes)
        g1.s1 = ((uint32_t)D_ & 0xFFFFu) << 16;        // tensor_dim0[15:0] @ bits 63:48
        g1.s2 = ((uint32_t)L_ & 0xFFFFu) << 16;        // tensor_dim1[15:0] @ bits 95:80
        g1.s3 = ((uint32_t)D_ & 0xFFFFu) << 16;        // tile_dim0 @ bits 127:112
        g1.s4 = (uint32_t)L_;                          // tile_dim1 @ bits 143:128
        g1.s5 = (uint32_t)D_;                          // tensor_dim0_stride[31:0]
        g1.s6 = 0u;                                    // stride0 hi / dim1_stride lo
        g1.s7 = 0u;

        asm volatile("tensor_load_to_lds %0, %1"
                     :
                     : "s"(g0), "s"(g1)
                     : "memory");
        __builtin_amdgcn_s_wait_tensorcnt(0);
    }
    __syncthreads();                                   // tile visible to all waves

    const int wave = t >> 5;
    const int lane = t & 31;
    const int h    = lane >> 4;                        // half-wave select
    const int l16  = lane & 15;
    const int rowBase = blockIdx.x * 128 + wave * 16;

    // A fragment address: lanes 0-15 carry K = k0,k0+1 ; lanes 16-31 K = k0+2,k0+3
    const float* aPtr = emb + ((size_t)b * N_ + rowBase + l16) * D_ + 2 * h;
    __builtin_prefetch(aPtr, 0, 1);                    // global_prefetch_b8

    v8f acc[4];
#pragma unroll
    for (int tt = 0; tt < 4; ++tt)
#pragma unroll
        for (int e = 0; e < 8; ++e) acc[tt][e] = 0.f;

    const float* bPtr0 = lmS + (size_t)(0 * 16 + l16) * D_ + 2 * h;
    const float* bPtr1 = lmS + (size_t)(1 * 16 + l16) * D_ + 2 * h;
    const float* bPtr2 = lmS + (size_t)(2 * 16 + l16) * D_ + 2 * h;
    const float* bPtr3 = lmS + (size_t)(3 * 16 + l16) * D_ + 2 * h;

    for (int k0 = 0; k0 < D_; k0 += 4) {
        float2 af = *(const float2*)(aPtr + k0);
        v2f a = {af.x, af.y};
        {
            float2 bf = *(const float2*)(bPtr0 + k0);
            v2f bb = {bf.x, bf.y};
            acc[0] = __builtin_amdgcn_wmma_f32_16x16x4_f32(false, a, false, bb,
                                                           (short)0, acc[0], false, false);
        }
        {
            float2 bf = *(const float2*)(bPtr1 + k0);
            v2f bb = {bf.x, bf.y};
            acc[1] = __builtin_amdgcn_wmma_f32_16x16x4_f32(false, a, false, bb,
                                                           (short)0, acc[1], false, false);
        }
        {
            float2 bf = *(const float2*)(bPtr2 + k0);
            v2f bb = {bf.x, bf.y};
            acc[2] = __builtin_amdgcn_wmma_f32_16x16x4_f32(false, a, false, bb,
                                                           (short)0, acc[2], false, false);
        }
        {
            float2 bf = *(const float2*)(bPtr3 + k0);
            v2f bb = {bf.x, bf.y};
            acc[3] = __builtin_amdgcn_wmma_f32_16x16x4_f32(false, a, false, bb,
                                                           (short)0, acc[3], false, false);
        }
    }

    // C/D layout: VGPR v, lanes 0-15 -> row v ; lanes 16-31 -> row v+8.
    float xr[8];
#pragma unroll
    for (int v = 0; v < 8; ++v)
        xr[v] = xn[(size_t)b * N_ + rowBase + v + 8 * h];
    float lt[4];
#pragma unroll
    for (int tt = 0; tt < 4; ++tt)
        lt[tt] = lnw[b * L_ + tt * 16 + l16];

#pragma unroll
    for (int v = 0; v < 8; ++v) {
        // local top-2 over this lane's 4 columns, lexicographic (dist, idx)
        float d0 = __builtin_inff(), d1 = __builtin_inff();
        int   i0 = L_, i1 = L_;
#pragma unroll
        for (int tt = 0; tt < 4; ++tt) {
            const float dot  = acc[tt][v];
            const float s    = xr[v] + lt[tt];
            const float diff = fmaxf(s - 2.f * dot, 1e-10f);
            const float fac  = fmaxf(fmaf(c, s, 1.0f), 1e-6f);
            const float dd   = sqrtf(diff) * sqrtf(fac);
            const int col    = tt * 16 + l16;
            if (dd < d0 || (dd == d0 && col < i0)) { d1 = d0; i1 = i0; d0 = dd; i0 = col; }
            else if (dd < d1 || (dd == d1 && col < i1)) { d1 = dd; i1 = col; }
        }
        // butterfly merge across the 16 lanes of this half-wave
#pragma unroll
        for (int m = 1; m < 16; m <<= 1) {
            float e0 = __shfl_xor(d0, m, 32); int j0 = __shfl_xor(i0, m, 32);
            float e1 = __shfl_xor(d1, m, 32); int j1 = __shfl_xor(i1, m, 32);
            float f0, f1; int g0, g1;
            const bool aFirst = (d0 < e0) || (d0 == e0 && i0 < j0);
            if (aFirst) {
                f0 = d0; g0 = i0;
                const bool sec = (d1 < e0) || (d1 == e0 && i1 < j0);
                f1 = sec ? d1 : e0; g1 = sec ? i1 : j0;
            } else {
                f0 = e0; g0 = j0;
                const bool sec = (e1 < d0) || (e1 == d0 && j1 < i0);
                f1 = sec ? e1 : d0; g1 = sec ? j1 : i0;
            }
            d0 = f0; i0 = g0; d1 = f1; i1 = g1;
        }
        if (l16 == 0) {
            const int i = min(i0, i1), j = max(i0, i1);
            // witness distance = 2nd-nearest; positive floats order-preserve as uint
            atomicMin(&ew[(b * L_ + i) * L_ + j], __float_as_uint(d1));
        }
    }
}

// ---------------------------------------------------------------------------
// Kernel 5: symmetrize -> adjacency -> connected components (min-label
// propagation, 64 iters) -> beta0/beta1/fragmentation/circular outputs.
// ---------------------------------------------------------------------------
__global__ __launch_bounds__(64) void k_final(const uint32_t* __restrict__ ew,
                                              float* __restrict__ out) {
    const int b = blockIdx.x;
    const int i = threadIdx.x;
    __shared__ int lbl[L_];
    __shared__ int red[L_];

    uint64_t mask = 0;
    for (int j = 0; j < L_; ++j) {
        const uint32_t a  = ew[(b * L_ + i) * L_ + j];
        const uint32_t bt = ew[(b * L_ + j) * L_ + i];
        const uint32_t m  = a < bt ? a : bt;
        if (m < 0x7F800000u) mask |= (1ull << j);      // finite after symmetrize
    }
    lbl[i] = i;
    __syncthreads();
    for (int it = 0; it < L_; ++it) {
        int nb = lbl[i];
        for (int j = 0; j < L_; ++j)
            if ((mask >> j) & 1) { const int lj = lbl[j]; nb = lj < nb ? lj : nb; }
        __syncthreads();
        lbl[i] = nb;
        __syncthreads();
    }
    const int isroot = (lbl[i] == i) ? 1 : 0;
    const uint64_t upper = mask & ~((2ull << i) - 1ull);   // bits j > i
    const int ecnt = __popcll(upper);

    red[i] = isroot; __syncthreads();
    for (int s = 32; s > 0; s >>= 1) { if (i < s) red[i] += red[i + s]; __syncthreads(); }
    const int ncomp = red[0]; __syncthreads();
    red[i] = ecnt; __syncthreads();
    for (int s = 32; s > 0; s >>= 1) { if (i < s) red[i] += red[i + s]; __syncthreads(); }
    const int nedges = red[0];

    if (i == 0) {
        const float beta0 = (float)ncomp;
        const float beta1 = (float)(nedges - (L_ - ncomp));
        out[b]      = beta0;
        out[8 + b]  = beta1;
        out[16 + b] = beta0 / (float)L_;
        out[24 + b] = (beta1 > 3.0f) ? 1.0f : 0.0f;
    }
}

// ---------------------------------------------------------------------------
// Launch
// ---------------------------------------------------------------------------
extern "C" void kernel_launch(void* const* d_in, const int* in_sizes, int n_in,
                              void* d_out, int out_size, void* d_ws, size_t ws_size,
                              hipStream_t stream) {
    const float* emb = (const float*)d_in[0];          // (8,16384,256) f32
    const float* cs  = (const float*)d_in[1];          // scalar
    float* out = (float*)d_out;                        // 32 floats

    char* w = (char*)d_ws;
    float*    xn     = (float*)(w);                    // B*N           (512 KB)
    float*    lmvecs = (float*)(w + 524288);           // B*L*D         (512 KB)
    float*    lnw    = (float*)(w + 1048576);          // B*L           (2 KB)
    int*      idxArr = (int*)  (w + 1048576 + 4096);   // B*L           (2 KB)
    uint32_t* ew     = (uint32_t*)(w + 1048576 + 8192);// B*L*L         (128 KB)

    k_rownorm<<<B_ * N_ / 256, 256, 0, stream>>>(emb, xn);
    k_init_ew<<<(B_ * L_ * L_ + 255) / 256, 256, 0, stream>>>(ew);
    k_select<<<B_, 1024, 0, stream>>>(emb, xn, cs, lmvecs, lnw, idxArr);
    dim3 g(N_ / 128, B_);
    k_witness<<<g, 256, 0, stream>>>(emb, xn, lmvecs, lnw, cs, ew);
    k_final<<<B_, 64, 0, stream>>>(ew, out);
}